// CustomHypergraphConv_2491081032063
// MI455X (gfx1250) — compile-verified
//
#include <hip/hip_runtime.h>

#define NUM_NODES   100000
#define NUM_EDGES   20000
#define NUM_INC     1600000
#define IN_CH       128
#define OUT_CH      128
#define EPS_F       1e-6f

typedef __attribute__((ext_vector_type(2))) float v2f;
typedef __attribute__((ext_vector_type(8))) float v8f;

// ---------------------------------------------------------------------------
// Kernel 1: degree counts (float, matches reference segment_sum of ones)
// ---------------------------------------------------------------------------
__global__ void hg_degrees(const int* __restrict__ node_idx,
                           const int* __restrict__ edge_idx,
                           float* __restrict__ deg_n,
                           float* __restrict__ deg_e)
{
    int i = blockIdx.x * blockDim.x + threadIdx.x;
    if (i < NUM_INC) {
        atomicAdd(&deg_n[node_idx[i]], 1.0f);
        atomicAdd(&deg_e[edge_idx[i]], 1.0f);
    }
}

// ---------------------------------------------------------------------------
// Kernel 2: x_t = x @ theta_w^T + theta_b  via V_WMMA_F32_16X16X4_F32
// One wave per 16-row M-tile; 8 N-tiles of 16; K-loop of 32 steps (K=4 each).
//
// A layout (16x4 f32, ISA 7.12.2): lanes 0-15 -> M=lane, VGPR{0,1}=K{0,1};
//   lanes 16-31 -> M=lane-16, VGPR{0,1}=K{2,3}.  -> contiguous float2 along K.
// B layout (4x16 f32): VGPR0 lanes[0-15]=row K0, lanes[16-31]=row K2;
//   VGPR1 = rows K1/K3. B[k][n] = theta_w[n][k] -> contiguous float2 along k.
// C/D layout (16x16 f32): VGPR r: lanes 0-15 -> M=r, N=lane;
//   lanes 16-31 -> M=r+8, N=lane-16.  Column fixed per lane -> bias broadcast.
// ---------------------------------------------------------------------------
__global__ void hg_gemm_bias_wmma(const float* __restrict__ x,
                                  const float* __restrict__ w,
                                  const float* __restrict__ bias,
                                  float* __restrict__ xt,
                                  int num_rows)
{
    const int wave  = threadIdx.x >> 5;
    const int lane  = threadIdx.x & 31;
    const int half  = lane >> 4;        // 0: lanes 0-15, 1: lanes 16-31
    const int nloc  = lane & 15;
    const int mtile = blockIdx.x * (blockDim.x >> 5) + wave;
    if (mtile * 16 >= num_rows) return;           // whole wave exits -> EXEC stays all-1
    const int m0 = mtile * 16;

    const float* __restrict__ arow = x + (size_t)(m0 + nloc) * IN_CH;

    #pragma unroll
    for (int nt = 0; nt < OUT_CH / 16; ++nt) {
        const int   n    = nt * 16 + nloc;
        const float bval = bias[n];
        v8f c = { bval, bval, bval, bval, bval, bval, bval, bval };
        const float* __restrict__ brow = w + (size_t)n * IN_CH;

        #pragma unroll
        for (int kt = 0; kt < IN_CH / 4; ++kt) {
            const int kb = kt * 4 + half * 2;
            v2f a, b;
            a.x = arow[kb];  a.y = arow[kb + 1];
            b.x = brow[kb];  b.y = brow[kb + 1];
            c = __builtin_amdgcn_wmma_f32_16x16x4_f32(
                    /*neg_a=*/false, a, /*neg_b=*/false, b,
                    /*c_mod=*/(short)0, c, /*reuse_a=*/false, /*reuse_b=*/false);
        }

        #pragma unroll
        for (int r = 0; r < 8; ++r) {
            const int row = m0 + (half ? r + 8 : r);
            xt[(size_t)row * OUT_CH + n] = c[r];
        }
    }
}

// ---------------------------------------------------------------------------
// Kernel 3: edge_feat[edge] += x_t[node]   (one wave per incidence entry;
// each lane moves a float4 = 4 channels, 4 f32 global atomics -> L2)
// ---------------------------------------------------------------------------
__global__ void hg_scatter_n2e(const float* __restrict__ xt,
                               const int* __restrict__ node_idx,
                               const int* __restrict__ edge_idx,
                               float* __restrict__ ef)
{
    const int inc  = blockIdx.x * (blockDim.x >> 5) + (threadIdx.x >> 5);
    const int lane = threadIdx.x & 31;
    if (inc >= NUM_INC) return;
    const int n = node_idx[inc];
    const int m = edge_idx[inc];
    const float4 v = ((const float4*)(xt + (size_t)n * OUT_CH))[lane];
    float* d = ef + (size_t)m * OUT_CH + lane * 4;
    atomicAdd(d + 0, v.x);
    atomicAdd(d + 1, v.y);
    atomicAdd(d + 2, v.z);
    atomicAdd(d + 3, v.w);
}

// ---------------------------------------------------------------------------
// Kernel 4: edge_feat *= hyperedge_weight * 1/(deg_e + eps)
// ---------------------------------------------------------------------------
__global__ void hg_scale_edges(float* __restrict__ ef,
                               const float* __restrict__ deg_e,
                               const float* __restrict__ hw)
{
    int i = blockIdx.x * blockDim.x + threadIdx.x;
    if (i < NUM_EDGES * OUT_CH) {
        int m = i >> 7;                 // /128
        ef[i] *= hw[m] * (1.0f / (deg_e[m] + EPS_F));
    }
}

// ---------------------------------------------------------------------------
// Kernel 5: out[node] += edge_feat[edge]
// ---------------------------------------------------------------------------
__global__ void hg_scatter_e2n(const float* __restrict__ ef,
                               const int* __restrict__ node_idx,
                               const int* __restrict__ edge_idx,
                               float* __restrict__ out)
{
    const int inc  = blockIdx.x * (blockDim.x >> 5) + (threadIdx.x >> 5);
    const int lane = threadIdx.x & 31;
    if (inc >= NUM_INC) return;
    const int n = node_idx[inc];
    const int m = edge_idx[inc];
    const float4 v = ((const float4*)(ef + (size_t)m * OUT_CH))[lane];
    float* d = out + (size_t)n * OUT_CH + lane * 4;
    atomicAdd(d + 0, v.x);
    atomicAdd(d + 1, v.y);
    atomicAdd(d + 2, v.z);
    atomicAdd(d + 3, v.w);
}

// ---------------------------------------------------------------------------
// Kernel 6: out *= 1/(deg_n + eps)
// ---------------------------------------------------------------------------
__global__ void hg_scale_nodes(float* __restrict__ out,
                               const float* __restrict__ deg_n)
{
    int i = blockIdx.x * blockDim.x + threadIdx.x;
    if (i < NUM_NODES * OUT_CH) {
        int n = i >> 7;
        out[i] *= 1.0f / (deg_n[n] + EPS_F);
    }
}

// ---------------------------------------------------------------------------
// Launch
// ---------------------------------------------------------------------------
extern "C" void kernel_launch(void* const* d_in, const int* in_sizes, int n_in,
                              void* d_out, int out_size, void* d_ws, size_t ws_size,
                              hipStream_t stream)
{
    const float* x        = (const float*)d_in[0];               // [N,128]
    const int*   hei      = (const int*)d_in[1];                 // [2,E]
    const float* theta_w  = (const float*)d_in[2];               // [128,128]
    const float* theta_b  = (const float*)d_in[3];               // [128]
    const float* hw       = (const float*)d_in[4];               // [M]
    float*       out      = (float*)d_out;                       // [N,128]

    const int* node_idx = hei;                 // row 0
    const int* edge_idx = hei + NUM_INC;       // row 1

    // workspace layout
    float* xt    = (float*)d_ws;                               // N*128 floats
    float* ef    = xt + (size_t)NUM_NODES * OUT_CH;            // M*128 floats
    float* deg_n = ef + (size_t)NUM_EDGES * OUT_CH;            // N floats
    float* deg_e = deg_n + NUM_NODES;                          // M floats

    // zero the atomic-accumulated buffers (capture-safe, stream-ordered)
    hipMemsetAsync(ef,    0, (size_t)NUM_EDGES * OUT_CH * sizeof(float), stream);
    hipMemsetAsync(deg_n, 0, (size_t)NUM_NODES * sizeof(float), stream);
    hipMemsetAsync(deg_e, 0, (size_t)NUM_EDGES * sizeof(float), stream);
    hipMemsetAsync(out,   0, (size_t)NUM_NODES * OUT_CH * sizeof(float), stream);

    // 1) degrees
    hg_degrees<<<(NUM_INC + 255) / 256, 256, 0, stream>>>(node_idx, edge_idx, deg_n, deg_e);

    // 2) x_t = x @ theta_w^T + b  (WMMA f32): 6250 M-tiles, 8 waves/block
    {
        const int mtiles = NUM_NODES / 16;                      // 6250
        const int waves_per_block = 8;                          // 256 threads
        const int blocks = (mtiles + waves_per_block - 1) / waves_per_block;
        hg_gemm_bias_wmma<<<blocks, 256, 0, stream>>>(x, theta_w, theta_b, xt, NUM_NODES);
    }

    // 3) scatter nodes -> edges (1 wave / incidence, 8 incidences / block)
    hg_scatter_n2e<<<NUM_INC / 8, 256, 0, stream>>>(xt, node_idx, edge_idx, ef);

    // 4) scale edges
    hg_scale_edges<<<(NUM_EDGES * OUT_CH + 255) / 256, 256, 0, stream>>>(ef, deg_e, hw);

    // 5) scatter edges -> nodes
    hg_scatter_e2n<<<NUM_INC / 8, 256, 0, stream>>>(ef, node_idx, edge_idx, out);

    // 6) scale nodes
    hg_scale_nodes<<<(NUM_NODES * OUT_CH + 255) / 256, 256, 0, stream>>>(out, deg_n);
}